// QuanvolutionClassifier_65481071409330
// MI455X (gfx1250) — compile-verified
//
#include <hip/hip_runtime.h>

typedef __attribute__((ext_vector_type(2))) float v2f;
typedef __attribute__((ext_vector_type(4))) float v4f;
typedef __attribute__((ext_vector_type(8))) float v8f;

#define ROWS_PER_BLOCK 64   // 4 waves * 16 batch rows
#define BLOCK_THREADS  128
#define FSTRIDE        58   // even pad -> 8B-aligned b64 LDS ops
#define KTOT           784
#define CHUNK_K        56   // one 2x28 image row-pair per chunk
#define NCHUNK         14
#define NV4            7    // 7 float4 x-loads per thread per chunk
#define NW             7    // 7 W floats per thread per chunk

__global__ __launch_bounds__(BLOCK_THREADS) void quanv_fused_kernel(
    const float* __restrict__ x,      // (B, 28, 28)
    const float* __restrict__ theta,  // (3, 4)
    const float* __restrict__ W,      // (10, 784)
    const float* __restrict__ bias,   // (10,)
    float* __restrict__ out,          // (B, 10) log-softmax
    int Bn)
{
    __shared__ float featsLDS[2][ROWS_PER_BLOCK * FSTRIDE]; // cos-features, k-permuted
    __shared__ float WtLDS[2][CHUNK_K * 16];                // Wt[k][n], cols 10..15 = 0

    const int tid  = threadIdx.x;
    const int lane = tid & 31;
    const int wave = tid >> 5;
    const int l    = lane & 15;   // A: row M ; B/C/D: col N
    const int half = lane >> 4;   // K-half select for A/B fragments

    const int b0 = blockIdx.x * ROWS_PER_BLOCK;

    // x staging: thread owns image row (2r + ro) of batch row b0+bl
    const int  bl  = tid >> 1;           // 0..63
    const int  ro  = tid & 1;            // 0/1
    const bool rowvalid = (b0 + bl) < Bn;
    const float* xrow = x + (size_t)(b0 + bl) * KTOT + ro * 28;

    // W staging: thread owns (n = tid>>3, k = tid&7 + 8j)
    const int  wn  = tid >> 3;           // 0..15
    const int  wkg = tid & 7;            // 0..7

    // Theta_w = sum over the 3 variational layers (RY angles compose additively).
    // This thread only ever touches wires {2*ro, 2*ro+1}.
    const float thA = ro ? (theta[2] + theta[6] + theta[10])
                         : (theta[0] + theta[4] + theta[8]);
    const float thB = ro ? (theta[3] + theta[7] + theta[11])
                         : (theta[1] + theta[5] + theta[9]);

    // --- helpers -----------------------------------------------------------
    auto load_chunk = [&](int r, v4f vx[NV4], float vw[NW]) {
        #pragma unroll
        for (int j = 0; j < NV4; ++j) {
            vx[j] = (v4f){0.f, 0.f, 0.f, 0.f};
            if (rowvalid) vx[j] = *(const v4f*)(xrow + r * CHUNK_K + j * 4);
        }
        #pragma unroll
        for (int j = 0; j < NW; ++j)
            vw[j] = (wn < 10) ? W[wn * KTOT + r * CHUNK_K + wkg + 8 * j] : 0.0f;
    };
    auto store_chunk = [&](int buf, const v4f vx[NV4], const float vw[NW]) {
        const int fbase = bl * FSTRIDE + 2 * ro;   // + 8*j4 + {0,1,4,5}
        #pragma unroll
        for (int j = 0; j < NV4; ++j) {
            v2f p0 = { __cosf(vx[j].x + thA), __cosf(vx[j].y + thB) };
            v2f p1 = { __cosf(vx[j].z + thA), __cosf(vx[j].w + thB) };
            *(v2f*)&featsLDS[buf][fbase + 8 * j]     = p0;
            *(v2f*)&featsLDS[buf][fbase + 8 * j + 4] = p1;
        }
        #pragma unroll
        for (int j = 0; j < NW; ++j)
            WtLDS[buf][(wkg + 8 * j) * 16 + wn] = vw[j];
    };
    // -----------------------------------------------------------------------

    v8f acc = {};  // 16x16 f32 accumulator tile
    v4f vx[NV4];
    float vw[NW];

    // prologue: stage chunk 0 into buffer 0
    load_chunk(0, vx, vw);
    store_chunk(0, vx, vw);
    __syncthreads();

    // pipelined loop: issue loads r+1 -> WMMA r -> encode+store r+1 -> barrier
    const int mrow = wave * 16 + l;
    for (int r = 0; r < NCHUNK; ++r) {
        const int cur = r & 1;
        const int nxt = cur ^ 1;
        const bool have_next = (r + 1 < NCHUNK);   // uniform

        if (have_next) load_chunk(r + 1, vx, vw);  // in flight during WMMAs

        #pragma unroll
        for (int kk = 0; kk < CHUNK_K; kk += 4) {
            const int k0 = kk + 2 * half;
            v2f a = *(const v2f*)&featsLDS[cur][mrow * FSTRIDE + k0]; // A[M=l,K=k0..k0+1]
            v2f bb;
            bb.x = WtLDS[cur][k0 * 16 + l];                          // B[K=k0,   N=l]
            bb.y = WtLDS[cur][(k0 + 1) * 16 + l];                    // B[K=k0+1, N=l]
            acc = __builtin_amdgcn_wmma_f32_16x16x4_f32(
                false, a, false, bb, (short)0, acc, false, false);
        }

        if (have_next) store_chunk(nxt, vx, vw);
        __syncthreads();  // next-buffer stores visible; current-buffer reads done
    }

    // epilogue: bias + log-softmax over the 10 valid columns.
    // C/D layout: lane holds column n = lane&15; VGPR i holds row i + 8*half.
    const int  n     = l;
    const bool valid = (n < 10);
    const float bn   = valid ? bias[n] : 0.0f;
    const int rowbase = b0 + wave * 16 + half * 8;
    const float NEG_INF = -__builtin_inff();

    #pragma unroll
    for (int i = 0; i < 8; ++i) {
        const float v = valid ? (acc[i] + bn) : NEG_INF;
        float m = v;
        #pragma unroll
        for (int off = 8; off >= 1; off >>= 1)
            m = fmaxf(m, __shfl_xor(m, off, 16));   // max within 16-lane group
        float s = valid ? expf(v - m) : 0.0f;
        #pragma unroll
        for (int off = 8; off >= 1; off >>= 1)
            s += __shfl_xor(s, off, 16);            // sum-exp within group
        const float res = v - m - logf(s);
        const int row = rowbase + i;
        if (valid && row < Bn)
            out[(size_t)row * 10 + n] = res;
    }
}

extern "C" void kernel_launch(void* const* d_in, const int* in_sizes, int n_in,
                              void* d_out, int out_size, void* d_ws, size_t ws_size,
                              hipStream_t stream) {
    (void)n_in; (void)out_size; (void)d_ws; (void)ws_size;
    const float* x     = (const float*)d_in[0];  // (B,28,28)
    const float* theta = (const float*)d_in[1];  // (3,4)
    const float* W     = (const float*)d_in[2];  // (10,784)
    const float* bias  = (const float*)d_in[3];  // (10,)
    float* out = (float*)d_out;

    const int Bn = in_sizes[0] / KTOT;           // 8192
    const int grid = (Bn + ROWS_PER_BLOCK - 1) / ROWS_PER_BLOCK;
    hipLaunchKernelGGL(quanv_fused_kernel, dim3(grid), dim3(BLOCK_THREADS), 0, stream,
                       x, theta, W, bias, out, Bn);
}